// GAT_TimeSeriesLayer_32650341384937
// MI455X (gfx1250) — compile-verified
//
#include <hip/hip_runtime.h>
#include <hip/hip_bf16.h>

typedef __attribute__((ext_vector_type(2))) float v2f;
typedef __attribute__((ext_vector_type(8))) float v8f;
typedef __attribute__((ext_vector_type(4))) int   v4i;

#define BB 16
#define TOBS 12
#define TPRED 20
#define NNODE 256
#define FIN 16
#define HID 128
#define HEADS 4
#define FOUT 2
#define ROWS (BB*TOBS*NNODE)   /* 49152 */
#define MROWS (BB*NNODE)       /* 4096  */

// ---- CDNA5 async global->LDS path (guarded; falls back to sync staging) ----
#if defined(__has_builtin)
#  if __has_builtin(__builtin_amdgcn_global_load_async_to_lds_b128) && \
      __has_builtin(__builtin_amdgcn_s_wait_asynccnt)
#    define HAVE_ASYNC_LDS 1
#  endif
#endif
#ifndef HAVE_ASYNC_LDS
#  define HAVE_ASYNC_LDS 0
#endif

#if HAVE_ASYNC_LDS
// builtin signature (from clang diagnostic): arg0 = int4 __device__* (global),
// arg1 = LDS destination, then immediate offset + cpol.
#define GPTR(p) ((__attribute__((address_space(1))) v4i*)(uintptr_t)(p))
#define LPTR(p) ((__attribute__((address_space(3))) v4i*)(uintptr_t)(p))
#endif

__device__ __forceinline__ v8f wmma4(v2f a, v2f b, v8f c) {
  // V_WMMA_F32_16X16X4_F32 : D = A(16x4,f32) x B(4x16,f32) + C(16x16,f32)
  return __builtin_amdgcn_wmma_f32_16x16x4_f32(false, a, false, b, (short)0, c, false, false);
}
__device__ __forceinline__ float sigmoidf_(float x){ return 1.0f/(1.0f + __expf(-x)); }
__device__ __forceinline__ v8f splat8(float x){ return (v8f){x,x,x,x,x,x,x,x}; }

// ---------------------------------------------------------------------------
// Generic fp32 WMMA GEMM: C[M,N] = A[M,K] @ B[K,N] (+bias) (+PReLU).
// Block = 8 waves = 8 consecutive 16-row tiles sharing ONE 64-col strip, so
// B fragments are identical across waves: B is staged into a double-buffered
// [PK][64] LDS panel (async->LDS when available), reads are conflict-free.
// ---------------------------------------------------------------------------
template<int PK, bool PRELU>
__global__ __launch_bounds__(256) void gemm_wmma_kernel(
    const float* __restrict__ A, const float* __restrict__ B,
    const float* __restrict__ bias, float* __restrict__ C,
    int M, int N, int K, const float* __restrict__ pa)
{
  __shared__ float bpan[2][PK][64];
  const int tid  = threadIdx.x, w = tid >> 5, lane = tid & 31;
  const int half = lane >> 4, lm = lane & 15;
  const int nstrips = N >> 6;
  const int mb = blockIdx.x / nstrips;
  const int n0 = (blockIdx.x % nstrips) * 64;
  const int m0 = (mb*8 + w) * 16;
  const int panels = K / PK;

  v8f acc[4];
#pragma unroll
  for (int j = 0; j < 4; j++) {
    float bv = bias ? bias[n0 + j*16 + lm] : 0.0f;
    acc[j] = splat8(bv);
  }
  const float* Arow = A + (size_t)(m0 + lm) * K;

  // stage panel 0
  for (int ch = tid; ch < PK*16; ch += 256) {
    const int kr = ch >> 4, c4 = (ch & 15) * 4;
    const float* src = B + (size_t)kr * N + n0 + c4;
#if HAVE_ASYNC_LDS
    __builtin_amdgcn_global_load_async_to_lds_b128(GPTR(src), LPTR(&bpan[0][kr][c4]), 0, 0);
#else
    *(float4*)&bpan[0][kr][c4] = *(const float4*)src;
#endif
  }
#if HAVE_ASYNC_LDS
  __builtin_amdgcn_s_wait_asynccnt(0);
#endif
  __syncthreads();

  for (int p = 0; p < panels; p++) {
    const int cur = p & 1;
    if (p + 1 < panels) {                       // prefetch next panel
      const int k0 = (p + 1) * PK;
      for (int ch = tid; ch < PK*16; ch += 256) {
        const int kr = ch >> 4, c4 = (ch & 15) * 4;
        const float* src = B + (size_t)(k0 + kr) * N + n0 + c4;
#if HAVE_ASYNC_LDS
        __builtin_amdgcn_global_load_async_to_lds_b128(GPTR(src), LPTR(&bpan[cur^1][kr][c4]), 0, 0);
#else
        *(float4*)&bpan[cur^1][kr][c4] = *(const float4*)src;
#endif
      }
    }
    for (int kk = 0; kk < PK; kk += 4) {
      const int ka = p*PK + kk + 2*half;        // global k (A)
      const int kl = kk + 2*half;               // panel-local k (B)
      v2f a; a.x = Arow[ka]; a.y = Arow[ka + 1];
#pragma unroll
      for (int j = 0; j < 4; j++) {
        v2f b; b.x = bpan[cur][kl][j*16 + lm]; b.y = bpan[cur][kl + 1][j*16 + lm];
        acc[j] = wmma4(a, b, acc[j]);
      }
    }
#if HAVE_ASYNC_LDS
    __builtin_amdgcn_s_wait_asynccnt(0);
#endif
    __syncthreads();
  }

  const float alpha = PRELU ? *pa : 0.0f;
#pragma unroll
  for (int j = 0; j < 4; j++) {
#pragma unroll
    for (int v = 0; v < 8; v++) {
      float x = acc[j][v];
      if (PRELU) x = (x >= 0.0f) ? x : alpha * x;
      C[(size_t)(m0 + v + 8*half) * N + n0 + j*16 + lm] = x;
    }
  }
}

// ---------------------------------------------------------------------------
// 128x128 transpose (for @W^T via the normal-orientation GEMM)
// ---------------------------------------------------------------------------
__global__ __launch_bounds__(256) void transpose128_kernel(
    const float* __restrict__ src, float* __restrict__ dst)
{
  const int t = blockIdx.x * 256 + threadIdx.x;   // 16384
  const int r = t >> 7, c = t & 127;
  dst[(size_t)c * 128 + r] = src[t];
}

// ---------------------------------------------------------------------------
// Per-row attention scores: s[row,h]=xp[row,h,:]·a_src[h,:] (wave per row)
// ---------------------------------------------------------------------------
__global__ __launch_bounds__(256) void scores_kernel(
    const float* __restrict__ xp, const float* __restrict__ asrc,
    const float* __restrict__ adst, float* __restrict__ s, float* __restrict__ d,
    int heads)
{
  const int row  = (blockIdx.x * blockDim.x + threadIdx.x) >> 5;
  const int lane = threadIdx.x & 31;
  if (row >= ROWS) return;
  const float* xr = xp + (size_t)row * heads * HID;
  for (int h = 0; h < heads; h++) {
    float sv = 0.0f, dv = 0.0f;
#pragma unroll
    for (int q = 0; q < 4; q++) {
      const int e = lane * 4 + q;
      const float x = xr[h*HID + e];
      sv += x * asrc[h*HID + e];
      dv += x * adst[h*HID + e];
    }
    for (int o = 16; o >= 1; o >>= 1) { sv += __shfl_xor(sv, o, 32); dv += __shfl_xor(dv, o, 32); }
    if (lane == 0) { s[(size_t)row*heads + h] = sv; d[(size_t)row*heads + h] = dv; }
  }
}

// ---------------------------------------------------------------------------
// GAT attention, 2 row-tiles (32 target rows) per block:
//   e[i,j]=leakyrelu(d[i]+s[j],0.2) masked; softmax_j -> LDS;
//   out = PReLU(attn @ xp_head + bias). Each B fragment feeds 2 WMMAs.
// ---------------------------------------------------------------------------
__global__ __launch_bounds__(256) void gat_attn_kernel(
    const float* __restrict__ xp, const float* __restrict__ s,
    const float* __restrict__ d, const int* __restrict__ adj,
    const float* __restrict__ bias, float* __restrict__ out,
    int heads, const float* __restrict__ pa)
{
  __shared__ float s_sh[NNODE];
  __shared__ float d_sh[32];
  __shared__ float attn[32][NNODE + 1];     // +1 pad: conflict-free col reads

  const int bid = blockIdx.x;
  const int it2 = bid & 7;                  // 32-row tile index
  const int h   = (bid >> 3) % heads;
  const int bt  = bid / (8 * heads);
  const int ld  = heads * HID;
  const int tid = threadIdx.x, w = tid >> 5, lane = tid & 31;

  if (tid < NNODE) s_sh[tid] = s[(size_t)(bt*NNODE + tid)*heads + h];
  if (tid < 32)    d_sh[tid] = d[(size_t)(bt*NNODE + it2*32 + tid)*heads + h];
  __syncthreads();

#pragma unroll
  for (int r = 0; r < 4; r++) {             // each wave: 4 softmax rows
    const int i  = w*4 + r;
    const int ig = it2*32 + i;
    const float dv = d_sh[i];
    float ev[8]; float mx = -3.0e38f;
#pragma unroll
    for (int jj = 0; jj < 8; jj++) {
      const int j = jj*32 + lane;
      float e = dv + s_sh[j];
      e = (e >= 0.0f) ? e : 0.2f * e;
      const bool nb = (adj[((size_t)bt*NNODE + j)*NNODE + ig] != 0) || (j == ig);
      e = nb ? e : -1.0e9f;
      ev[jj] = e; mx = fmaxf(mx, e);
    }
    for (int o = 16; o >= 1; o >>= 1) mx = fmaxf(mx, __shfl_xor(mx, o, 32));
    float sum = 0.0f;
#pragma unroll
    for (int jj = 0; jj < 8; jj++) { ev[jj] = __expf(ev[jj] - mx); sum += ev[jj]; }
    for (int o = 16; o >= 1; o >>= 1) sum += __shfl_xor(sum, o, 32);
    const float inv = 1.0f / sum;
#pragma unroll
    for (int jj = 0; jj < 8; jj++) attn[i][jj*32 + lane] = ev[jj] * inv;
  }
  __syncthreads();

  // WMMA: wave w -> cols [w*16,w*16+16); two 16-row tiles share each B frag
  const int half = lane >> 4, lm = lane & 15;
  const int c0 = w * 16;
  const float bv = bias[h*HID + c0 + lm];
  v8f acc0 = splat8(bv), acc1 = splat8(bv);
  const float* xb = xp + (size_t)bt*NNODE*ld + h*HID + c0 + lm;
  for (int kk = 0; kk < NNODE; kk += 4) {
    const int ka = kk + 2*half;
    v2f a0; a0.x = attn[lm][ka];      a0.y = attn[lm][ka + 1];
    v2f a1; a1.x = attn[16 + lm][ka]; a1.y = attn[16 + lm][ka + 1];
    v2f b;  b.x  = xb[(size_t)ka * ld]; b.y = xb[(size_t)(ka+1) * ld];
    acc0 = wmma4(a0, b, acc0);
    acc1 = wmma4(a1, b, acc1);
  }
  const float alpha = *pa;
#pragma unroll
  for (int v = 0; v < 8; v++) {
    float x0 = acc0[v]; x0 = (x0 >= 0.0f) ? x0 : alpha * x0;
    float x1 = acc1[v]; x1 = (x1 >= 0.0f) ? x1 : alpha * x1;
    const size_t r0 = (size_t)bt*NNODE + it2*32 + v + 8*half;
    out[r0 * ld + h*HID + c0 + lm]        = x0;
    out[(r0 + 16) * ld + h*HID + c0 + lm] = x1;
  }
}

// ---------------------------------------------------------------------------
// Fused 2-layer GRU. Block owns 32 sequence rows (2 row-tiles per wave) so
// each weight fragment feeds 2 WMMAs. gi/gh r,z-gate GEMMs accumulate into a
// SHARED accumulator (pre-activations add); only the n-gate stays split.
// h round-trips through padded LDS to feed the next step's A-matrix.
// ---------------------------------------------------------------------------
__global__ __launch_bounds__(256) void gru_kernel(
    const float* __restrict__ x4,
    const float* __restrict__ wih0, const float* __restrict__ whh0,
    const float* __restrict__ bih0, const float* __restrict__ bhh0,
    const float* __restrict__ wih1, const float* __restrict__ whh1,
    const float* __restrict__ bih1, const float* __restrict__ bhh1,
    float* __restrict__ ys, float* __restrict__ out)
{
  __shared__ float hbuf[32][HID + 2];
  const int tid = threadIdx.x, w = tid >> 5, lane = tid & 31;
  const int half = lane >> 4, lm = lane & 15;
  const int m0 = blockIdx.x * 32;
  const int c  = w*16 + lm;                         // hidden column owned

  for (int i = tid; i < 32*(HID+2); i += 256) (&hbuf[0][0])[i] = 0.0f;
  __syncthreads();
  v8f h0 = splat8(0.0f), h1 = splat8(0.0f);

  for (int layer = 0; layer < 2; layer++) {
    const float* wih = layer ? wih1 : wih0;
    const float* whh = layer ? whh1 : whh0;
    const float* bih = layer ? bih1 : bih0;
    const float* bhh = layer ? bhh1 : bhh0;
    const float* xs  = layer ? ys  : x4;
    float*       dst = layer ? out : ys;

    for (int t = 0; t < TOBS; t++) {
      const float brz = bih[c] + bhh[c];
      const float bzz = bih[HID + c] + bhh[HID + c];
      v8f ar0 = splat8(brz), ar1 = splat8(brz);          // r-gate (gi+gh)
      v8f az0 = splat8(bzz), az1 = splat8(bzz);          // z-gate (gi+gh)
      v8f an0 = splat8(bih[2*HID + c]), an1 = an0;       // i-part of n-gate
      v8f hn0 = splat8(bhh[2*HID + c]), hn1 = hn0;       // h-part of n-gate

      const float* Ar0 = xs + ((size_t)(m0 + lm)*TOBS + t) * HID;
      const float* Ar1 = xs + ((size_t)(m0 + 16 + lm)*TOBS + t) * HID;
      for (int kk = 0; kk < HID; kk += 4) {
        const int ka = kk + 2*half;
        v2f a0; a0.x = Ar0[ka]; a0.y = Ar0[ka + 1];
        v2f a1; a1.x = Ar1[ka]; a1.y = Ar1[ka + 1];
        v2f br; br.x = wih[(size_t)(0*HID + c)*HID + ka]; br.y = wih[(size_t)(0*HID + c)*HID + ka + 1];
        v2f bz; bz.x = wih[(size_t)(1*HID + c)*HID + ka]; bz.y = wih[(size_t)(1*HID + c)*HID + ka + 1];
        v2f bn; bn.x = wih[(size_t)(2*HID + c)*HID + ka]; bn.y = wih[(size_t)(2*HID + c)*HID + ka + 1];
        ar0 = wmma4(a0, br, ar0); ar1 = wmma4(a1, br, ar1);
        az0 = wmma4(a0, bz, az0); az1 = wmma4(a1, bz, az1);
        an0 = wmma4(a0, bn, an0); an1 = wmma4(a1, bn, an1);
      }
      for (int kk = 0; kk < HID; kk += 4) {
        const int ka = kk + 2*half;
        v2f a0; a0.x = hbuf[lm][ka];      a0.y = hbuf[lm][ka + 1];
        v2f a1; a1.x = hbuf[16 + lm][ka]; a1.y = hbuf[16 + lm][ka + 1];
        v2f br; br.x = whh[(size_t)(0*HID + c)*HID + ka]; br.y = whh[(size_t)(0*HID + c)*HID + ka + 1];
        v2f bz; bz.x = whh[(size_t)(1*HID + c)*HID + ka]; bz.y = whh[(size_t)(1*HID + c)*HID + ka + 1];
        v2f bn; bn.x = whh[(size_t)(2*HID + c)*HID + ka]; bn.y = whh[(size_t)(2*HID + c)*HID + ka + 1];
        ar0 = wmma4(a0, br, ar0); ar1 = wmma4(a1, br, ar1);
        az0 = wmma4(a0, bz, az0); az1 = wmma4(a1, bz, az1);
        hn0 = wmma4(a0, bn, hn0); hn1 = wmma4(a1, bn, hn1);
      }
      v8f nh0, nh1;
#pragma unroll
      for (int v = 0; v < 8; v++) {
        {
          const float r  = sigmoidf_(ar0[v]);
          const float z  = sigmoidf_(az0[v]);
          const float nn = tanhf(an0[v] + r * hn0[v]);
          nh0[v] = (1.0f - z) * nn + z * h0[v];
        }
        {
          const float r  = sigmoidf_(ar1[v]);
          const float z  = sigmoidf_(az1[v]);
          const float nn = tanhf(an1[v] + r * hn1[v]);
          nh1[v] = (1.0f - z) * nn + z * h1[v];
        }
      }
      h0 = nh0; h1 = nh1;
      __syncthreads();                       // all gh reads of old h done
#pragma unroll
      for (int v = 0; v < 8; v++) {
        hbuf[v + 8*half][c]      = h0[v];
        hbuf[16 + v + 8*half][c] = h1[v];
        dst[((size_t)(m0 + v + 8*half)*TOBS + t) * HID + c]      = h0[v];
        dst[((size_t)(m0 + 16 + v + 8*half)*TOBS + t) * HID + c] = h1[v];
      }
      __syncthreads();                       // new h visible to all waves
    }
    if (layer == 0) {                        // reset state between layers
      h0 = splat8(0.0f); h1 = splat8(0.0f);
      for (int i = tid; i < 32*(HID+2); i += 256) (&hbuf[0][0])[i] = 0.0f;
      __syncthreads();
    }
  }
}

// ---------------------------------------------------------------------------
// 3x3 conv over (N,HID) spatial, 12 in-ch -> 20 out-ch, pad=1, +bias +PReLU.
// x5[b,to,n,c] = gru_out[((b*256+n)*12+to)*128+c].
// ---------------------------------------------------------------------------
__global__ __launch_bounds__(256) void conv_kernel(
    const float* __restrict__ g, const float* __restrict__ wt,
    const float* __restrict__ cb, float* __restrict__ x7,
    const float* __restrict__ pa)
{
  const int idx = blockIdx.x * 256 + threadIdx.x;     // < 16*20*256*128
  const int cc = idx & 127;
  const int n  = (idx >> 7) & 255;
  const int tp = (idx >> 15) % TPRED;
  const int b  = idx / (128 * 256 * TPRED);
  float acc = cb[tp];
  for (int to = 0; to < TOBS; to++) {
    const float* wp = wt + (size_t)(tp*TOBS + to) * 9;
    const float* gp = g + ((size_t)(b*NNODE + n)*TOBS + to) * HID + cc;
#pragma unroll
    for (int dh = 0; dh < 3; dh++) {
      const int nn = n + dh - 1;
      if (nn < 0 || nn >= NNODE) continue;
#pragma unroll
      for (int dw = 0; dw < 3; dw++) {
        const int c2 = cc + dw - 1;
        if (c2 < 0 || c2 >= HID) continue;
        acc += gp[(size_t)(dh - 1) * (TOBS*HID) + (dw - 1)] * wp[dh*3 + dw];
      }
    }
  }
  const float alpha = *pa;
  x7[idx] = (acc >= 0.0f) ? acc : alpha * acc;
}

// ---------------------------------------------------------------------------
// Final 128->2 head: block(128) per row, coalesced load + block reduction.
// ---------------------------------------------------------------------------
__global__ __launch_bounds__(128) void head2_kernel(
    const float* __restrict__ x9, const float* __restrict__ w2,
    const float* __restrict__ b2, float* __restrict__ out)
{
  __shared__ float r0[4], r1[4];
  const int row = blockIdx.x, tid = threadIdx.x;
  const float xv = x9[(size_t)row * HID + tid];
  float p0 = xv * w2[tid], p1 = xv * w2[HID + tid];
  for (int o = 16; o >= 1; o >>= 1) { p0 += __shfl_xor(p0, o, 32); p1 += __shfl_xor(p1, o, 32); }
  if ((tid & 31) == 0) { r0[tid >> 5] = p0; r1[tid >> 5] = p1; }
  __syncthreads();
  if (tid == 0) {
    out[(size_t)row*2 + 0] = r0[0]+r0[1]+r0[2]+r0[3] + b2[0];
    out[(size_t)row*2 + 1] = r1[0]+r1[1]+r1[2]+r1[3] + b2[1];
  }
}

// ---------------------------------------------------------------------------
extern "C" void kernel_launch(void* const* d_in, const int* in_sizes, int n_in,
                              void* d_out, int out_size, void* d_ws, size_t ws_size,
                              hipStream_t stream) {
  const float* x       = (const float*)d_in[0];
  const int*   adj     = (const int*)  d_in[1];
  const float* W1      = (const float*)d_in[2];
  const float* a_src1  = (const float*)d_in[3];
  const float* a_dst1  = (const float*)d_in[4];
  const float* b1      = (const float*)d_in[5];
  const float* W2      = (const float*)d_in[6];
  const float* a_src2  = (const float*)d_in[7];
  const float* a_dst2  = (const float*)d_in[8];
  const float* b2      = (const float*)d_in[9];
  const float* wih0    = (const float*)d_in[10];
  const float* whh0    = (const float*)d_in[11];
  const float* bih0    = (const float*)d_in[12];
  const float* bhh0    = (const float*)d_in[13];
  const float* wih1    = (const float*)d_in[14];
  const float* whh1    = (const float*)d_in[15];
  const float* bih1    = (const float*)d_in[16];
  const float* bhh1    = (const float*)d_in[17];
  const float* out1_w  = (const float*)d_in[18];
  const float* out1_b  = (const float*)d_in[19];
  const float* out2_w  = (const float*)d_in[20];
  const float* out2_b  = (const float*)d_in[21];
  const float* conv_w  = (const float*)d_in[22];
  const float* conv_b  = (const float*)d_in[23];
  const float* pa      = (const float*)d_in[24];

  float* ws = (float*)d_ws;
  const size_t R = (size_t)ROWS;
  float* XP1 = ws;                         // ROWS*512
  float* X2  = XP1 + R*512;                // ROWS*512
  float* XP2 = X2  + R*512;                // ROWS*128
  float* X4  = XP2 + R*128;                // ROWS*128
  float* GRU = X4  + R*128;                // ROWS*128 (layer-1 output)
  float* YS  = GRU + R*128;                // ROWS*128 (layer-0 output)
  float* S1  = YS  + R*128;                // ROWS*4
  float* D1  = S1  + R*4;
  float* S2  = D1  + R*4;                  // ROWS
  float* D2  = S2  + R;
  float* WT  = D2  + R;                    // 128*128 (out1_w^T)
  float* X7  = XP1;                        // reuse (81920*128 <= ROWS*512)
  float* X9  = X2;                         // reuse

  // 0) out1_w^T (independent; overlaps the front of the pipeline)
  transpose128_kernel<<<64, 256, 0, stream>>>(out1_w, WT);
  // 1) xp1 = x @ W1   [49152x16]@[16x512]
  gemm_wmma_kernel<16,false><<<(ROWS/128)*(512/64), 256, 0, stream>>>(
      x, W1, nullptr, XP1, ROWS, 512, FIN, pa);
  // 2) per-head attention scores
  scores_kernel<<<ROWS/8, 256, 0, stream>>>(XP1, a_src1, a_dst1, S1, D1, HEADS);
  // 3) GAT1 attention + bias + PReLU -> x2
  gat_attn_kernel<<<BB*TOBS*HEADS*8, 256, 0, stream>>>(XP1, S1, D1, adj, b1, X2, HEADS, pa);
  // 4) xp2 = x2 @ W2  [49152x512]@[512x128]
  gemm_wmma_kernel<32,false><<<(ROWS/128)*(128/64), 256, 0, stream>>>(
      X2, W2, nullptr, XP2, ROWS, 128, 512, pa);
  // 5) scores (1 head)
  scores_kernel<<<ROWS/8, 256, 0, stream>>>(XP2, a_src2, a_dst2, S2, D2, 1);
  // 6) GAT2 attention + bias + PReLU -> x4
  gat_attn_kernel<<<BB*TOBS*1*8, 256, 0, stream>>>(XP2, S2, D2, adj, b2, X4, 1, pa);
  // 7) fused 2-layer GRU over the raw-reshaped [4096,12,128] view
  gru_kernel<<<MROWS/32, 256, 0, stream>>>(X4, wih0, whh0, bih0, bhh0,
                                           wih1, whh1, bih1, bhh1, YS, GRU);
  // 8) 3x3 conv (12ch->20ch over [256,128]) + bias + PReLU -> x7
  conv_kernel<<<(BB*TPRED*NNODE*HID)/256, 256, 0, stream>>>(GRU, conv_w, conv_b, X7, pa);
  // 9) x9 = PReLU(x7 @ out1_w^T + out1_b) via transposed weights
  gemm_wmma_kernel<32,true><<<((BB*TPRED*NNODE)/128)*(128/64), 256, 0, stream>>>(
      X7, WT, out1_b, X9, BB*TPRED*NNODE, HID, HID, pa);
  // 10) out = x9 @ out2_w^T + out2_b  -> [16,20,256,2]
  head2_kernel<<<BB*TPRED*NNODE, 128, 0, stream>>>(X9, out2_w, out2_b, (float*)d_out);
}